// LODAttention_30588757082705
// MI455X (gfx1250) — compile-verified
//
#include <hip/hip_runtime.h>
#include <math.h>

// ---------------------------------------------------------------------------
// LOD attention for gfx1250 (MI455X), fp32 end-to-end via V_WMMA_F32_16X16X4_F32.
// Shapes: B=1, S=2048, D_MODEL=1024, H=16, D=64, G=32, WIN=512, top1=8.
// LOD2 level is a no-op (top-2-of-2 selects all chunks, identity order), so:
// QKV GEMM -> RoPE -> k1 proj -> per-row top8 chunk mask -> block-sparse
// causal flash attention -> output GEMM.
//
// The big GEMMs use the CDNA5 Tensor Data Mover (TENSOR_LOAD_TO_LDS +
// s_wait_tensorcnt) to DMA A/W tiles into LDS with double buffering, so the
// WMMA pipe reads ds_load_b64 fragments instead of stalling on
// s_wait_loadcnt 0x0 behind global loads. LDS rows are padded to 34 floats
// via the TDM pad feature -> conflict-free b64 fragment reads.
// This toolchain exposes the 6-arg tensor_load_to_lds (trailing i32x8 group
// is zero for 2-D tiles), per the round-3 diagnostic.
// ---------------------------------------------------------------------------

typedef float v2f __attribute__((ext_vector_type(2)));
typedef float v8f __attribute__((ext_vector_type(8)));
typedef unsigned int u32x4 __attribute__((ext_vector_type(4)));
typedef int i32x8 __attribute__((ext_vector_type(8)));
typedef int i32x4 __attribute__((ext_vector_type(4)));

#define SCALE_F 0.125f      // D^-0.5, D=64
#define NEG_BIG (-3.0e38f)  // masked-score sentinel (finite: avoids inf-inf NaN)

#if __has_builtin(__builtin_amdgcn_tensor_load_to_lds) && \
    __has_builtin(__builtin_amdgcn_s_wait_tensorcnt)
#define USE_TDM 1
#else
#define USE_TDM 0
#endif

__device__ __forceinline__ v8f wmma4(v2f a, v2f b, v8f c) {
  // D(16x16,f32) = A(16x4,f32) * B(4x16,f32) + C
  return __builtin_amdgcn_wmma_f32_16x16x4_f32(
      /*neg_a=*/false, a, /*neg_b=*/false, b,
      /*c_mod=*/(short)0, c, /*reuse_a=*/false, /*reuse_b=*/false);
}

__device__ __forceinline__ v2f ld2(const float* p) { return *(const v2f*)p; }

#if USE_TDM
// ---------------------------------------------------------------------------
// TDM: DMA a 64(rows) x 32(cols) f32 tile, row stride = kstride elements,
// from global `gsrc` into LDS at byte offset `lds_off`, with 2 DWORDs of pad
// appended every 32 DWORDs (row pitch in LDS = 34 floats).
// D# per CDNA5 ISA ch.8: group0 {count=1, lds_addr, global_addr, type=2},
// group1 {data_size=2(4B), pad_enable, pad_interval=4(32dw), pad_amount=1(2dw),
//         tensor_dim0=kstride, tensor_dim1=1<<20, tile_dim0=32, tile_dim1=64,
//         tensor_dim0_stride=kstride}, groups 2/3 (and trailing group) = 0.
// ---------------------------------------------------------------------------
__device__ __forceinline__ void tdm_load_tile_64x32(const float* gsrc,
                                                    unsigned lds_off,
                                                    unsigned kstride) {
  unsigned long long ga = (unsigned long long)gsrc;
  u32x4 g0 = { 1u,                                  // count=1 (user descriptor)
               lds_off,                             // LDS byte address
               (unsigned)ga,                        // global addr [31:0]
               (unsigned)(ga >> 32) | (2u << 30) }; // global addr [56:32] | type=2
  i32x8 g1 = {
    (int)((2u << 16) | (1u << 20) | (4u << 22) | (1u << 25)),
      // data_size=4B | pad_enable | pad_interval: 32 dwords | pad_amount: 2 dwords
    (int)((kstride & 0xFFFFu) << 16),        // tensor_dim0[15:0]  (abar addr = 0)
    (int)((kstride >> 16) & 0xFFFFu),        // tensor_dim0[31:16] | dim1[15:0]=0
    (int)(16u | (32u << 16)),                // dim1[31:16] (=1<<20) | tile_dim0=32
    (int)64u,                                // tile_dim1=64 | tile_dim2=0
    (int)kstride,                            // tensor_dim0_stride[31:0]
    0,                                       // stride hi | dim1_stride lo (unused 2D)
    0 };
  i32x4 g2 = { 0, 0, 0, 0 };
  i32x4 g3 = { 0, 0, 0, 0 };
  i32x8 g4 = { 0, 0, 0, 0, 0, 0, 0, 0 };     // trailing group (6-arg form), unused 2D
  __builtin_amdgcn_tensor_load_to_lds(g0, g1, g2, g3, g4, /*cpol=*/0);
}
#endif

// ---------------------------------------------------------------------------
// C(MxN) = A(MxK) @ W(NxK)^T    (row-major), M%64==0, N%64==0, K%32==0.
// Block = 4 waves -> 64x64 C tile; per-wave 16(M) x 64(N) (4 f32 accums).
// TDM path: K chunked by 32; A/W 64x32 tiles DMA'd to LDS (double-buffered,
// pitch 34), fragments read with ds_load_b64.
// ---------------------------------------------------------------------------
__global__ __launch_bounds__(128) void gemm_xwt(const float* __restrict__ A,
                                                const float* __restrict__ W,
                                                float* __restrict__ C,
                                                int K, int N) {
  const int lane = threadIdx.x & 31;
  const int wave = threadIdx.x >> 5;
  const int m0 = blockIdx.y * 64;
  const int n0 = blockIdx.x * 64;
  const int l15  = lane & 15;
  const int koff = (lane >> 4) << 1;   // 0 or 2
  const int hi8  = (lane >> 4) << 3;   // 0 or 8

  v8f acc0 = {}, acc1 = {}, acc2 = {}, acc3 = {};

#if USE_TDM
  __shared__ float As[2][64 * 34];     // pitch 34 = 32 data + 2 pad dwords
  __shared__ float Bs[2][64 * 34];

  const float* Ablk = A + (size_t)m0 * K;
  const float* Wblk = W + (size_t)n0 * K;
  const int NC = K >> 5;               // number of 32-wide K chunks

  if (threadIdx.x == 0) {              // TDM issues once per wave; wave0 only
    tdm_load_tile_64x32(Ablk, (unsigned)(size_t)&As[0][0], (unsigned)K);
    tdm_load_tile_64x32(Wblk, (unsigned)(size_t)&Bs[0][0], (unsigned)K);
    if (NC > 1) {
      tdm_load_tile_64x32(Ablk + 32, (unsigned)(size_t)&As[1][0], (unsigned)K);
      tdm_load_tile_64x32(Wblk + 32, (unsigned)(size_t)&Bs[1][0], (unsigned)K);
    }
  }

  for (int ic = 0; ic < NC; ++ic) {
    const int buf = ic & 1;
    if (threadIdx.x == 0) {
      if (ic + 1 < NC) __builtin_amdgcn_s_wait_tensorcnt(2);  // chunk ic done
      else             __builtin_amdgcn_s_wait_tensorcnt(0);
    }
    __syncthreads();

    const float* as  = &As[buf][(wave * 16 + l15) * 34 + koff];
    const float* bs0 = &Bs[buf][( 0 + l15) * 34 + koff];
    const float* bs1 = &Bs[buf][(16 + l15) * 34 + koff];
    const float* bs2 = &Bs[buf][(32 + l15) * 34 + koff];
    const float* bs3 = &Bs[buf][(48 + l15) * 34 + koff];
#pragma unroll
    for (int ks = 0; ks < 8; ++ks) {
      v2f a  = ld2(as  + ks * 4);
      v2f b0 = ld2(bs0 + ks * 4);
      v2f b1 = ld2(bs1 + ks * 4);
      v2f b2 = ld2(bs2 + ks * 4);
      v2f b3 = ld2(bs3 + ks * 4);
      acc0 = wmma4(a, b0, acc0);
      acc1 = wmma4(a, b1, acc1);
      acc2 = wmma4(a, b2, acc2);
      acc3 = wmma4(a, b3, acc3);
    }
    __syncthreads();                   // all reads of buf done before refill
    if (threadIdx.x == 0 && ic + 2 < NC) {
      tdm_load_tile_64x32(Ablk + (ic + 2) * 32,
                          (unsigned)(size_t)&As[buf][0], (unsigned)K);
      tdm_load_tile_64x32(Wblk + (ic + 2) * 32,
                          (unsigned)(size_t)&Bs[buf][0], (unsigned)K);
    }
  }
#else
  // Fallback: direct global-load pipeline (round-2 version).
  const float* arow = A + (size_t)(m0 + wave * 16 + l15) * K + koff;
  const float* w0 = W + (size_t)(n0 +  0 + l15) * K + koff;
  const float* w1 = W + (size_t)(n0 + 16 + l15) * K + koff;
  const float* w2 = W + (size_t)(n0 + 32 + l15) * K + koff;
  const float* w3 = W + (size_t)(n0 + 48 + l15) * K + koff;
  v2f a_c = ld2(arow);
  v2f b0c = ld2(w0), b1c = ld2(w1), b2c = ld2(w2), b3c = ld2(w3);
#pragma unroll 4
  for (int k0 = 4; k0 < K; k0 += 4) {
    v2f a_n = ld2(arow + k0);
    v2f b0n = ld2(w0 + k0);
    v2f b1n = ld2(w1 + k0);
    v2f b2n = ld2(w2 + k0);
    v2f b3n = ld2(w3 + k0);
    acc0 = wmma4(a_c, b0c, acc0);
    acc1 = wmma4(a_c, b1c, acc1);
    acc2 = wmma4(a_c, b2c, acc2);
    acc3 = wmma4(a_c, b3c, acc3);
    a_c = a_n; b0c = b0n; b1c = b1n; b2c = b2n; b3c = b3n;
  }
  acc0 = wmma4(a_c, b0c, acc0);
  acc1 = wmma4(a_c, b1c, acc1);
  acc2 = wmma4(a_c, b2c, acc2);
  acc3 = wmma4(a_c, b3c, acc3);
#endif

  float* crow = C + n0 + l15;
#pragma unroll
  for (int r = 0; r < 8; ++r) {
    size_t m = (size_t)(m0 + wave * 16 + r + hi8) * N;
    crow[m +  0] = acc0[r];
    crow[m + 16] = acc1[r];
    crow[m + 32] = acc2[r];
    crow[m + 48] = acc3[r];
  }
}

// ---------------------------------------------------------------------------
// RoPE applied in-place to Q and K, layout (S, H*D) = (2048, 1024).
// ---------------------------------------------------------------------------
__global__ __launch_bounds__(256) void rope_qk(float* __restrict__ Qb,
                                               float* __restrict__ Kb) {
  int idx = blockIdx.x * blockDim.x + threadIdx.x;   // 2048*16*32 threads
  int s = idx >> 9;
  int h = (idx >> 5) & 15;
  int d = idx & 31;
  float inv = powf(10000.0f, -(float)d * (1.0f / 32.0f));  // inv_freq[d]
  float ang = (float)s * inv;
  float cs = cosf(ang), sn = sinf(ang);
  int base = s * 1024 + h * 64 + d;
  float a0 = Qb[base], a1 = Qb[base + 32];
  Qb[base]      = a0 * cs - a1 * sn;
  Qb[base + 32] = a1 * cs + a0 * sn;
  float b0 = Kb[base], b1 = Kb[base + 32];
  Kb[base]      = b0 * cs - b1 * sn;
  Kb[base + 32] = b1 * cs + b0 * sn;
}

// ---------------------------------------------------------------------------
// k1[h, c, :] = (K chunk c of head h, flattened 32 tok x 64 dim = 2048)
//               @ k_lod1(64 x 2048)^T.  One block/head, 8 waves;
// each wave 16(M=chunks) x 32(N), pipelined K-loop over 2048.
// ---------------------------------------------------------------------------
__global__ __launch_bounds__(256) void k1_proj(const float* __restrict__ Kb,
                                               const float* __restrict__ lod1,
                                               float* __restrict__ K1) {
  const int h = blockIdx.x;
  const int lane = threadIdx.x & 31;
  const int wave = threadIdx.x >> 5;   // 0..7
  const int mt = wave >> 1;            // chunk tile 0..3
  const int nt = (wave & 1) * 2;       // n tile 0 or 2
  const int l15  = lane & 15;
  const int koff = (lane >> 4) << 1;
  const int hi8  = (lane >> 4) << 3;
  const int c = mt * 16 + l15;         // A row = chunk index

  v8f acc0 = {}, acc1 = {};
  const float* b0 = lod1 + (size_t)(nt * 16 + l15) * 2048 + koff;
  const float* b1 = lod1 + (size_t)((nt + 1) * 16 + l15) * 2048 + koff;
  const float* kh = Kb + h * 64;

  // preload k0 = 0
  v2f a_c = ld2(&kh[(size_t)(c * 32 + (koff >> 6)) * 1024 + (koff & 63)]);
  v2f p0 = ld2(b0), p1 = ld2(b1);

#pragma unroll 4
  for (int k0 = 4; k0 < 2048; k0 += 4) {
    int kk = k0 + koff;                // flattened (g,d); kk,kk+1 share g
    int g = kk >> 6, d = kk & 63;
    v2f a_n = ld2(&kh[(size_t)(c * 32 + g) * 1024 + d]);
    v2f n0 = ld2(b0 + k0);
    v2f n1 = ld2(b1 + k0);
    acc0 = wmma4(a_c, p0, acc0);
    acc1 = wmma4(a_c, p1, acc1);
    a_c = a_n; p0 = n0; p1 = n1;
  }
  acc0 = wmma4(a_c, p0, acc0);
  acc1 = wmma4(a_c, p1, acc1);

#pragma unroll
  for (int r = 0; r < 8; ++r) {
    int cc = mt * 16 + r + hi8;
    K1[h * 4096 + cc * 64 + nt * 16 + l15]       = acc0[r];
    K1[h * 4096 + cc * 64 + (nt + 1) * 16 + l15] = acc1[r];
  }
}

// ---------------------------------------------------------------------------
// scores1 = Q(16 rows) x k1(h)^T -> 16x64 via WMMA, then per-row top-8
// (strict-max, lowest index wins == jax.lax.top_k) -> 64-bit chunk mask.
// One wave per (q-tile, head).
// ---------------------------------------------------------------------------
__global__ __launch_bounds__(32) void lod_topk(const float* __restrict__ Q,
                                               const float* __restrict__ K1,
                                               unsigned long long* __restrict__ Msk) {
  const int qt = blockIdx.x, h = blockIdx.y;
  const int lane = threadIdx.x;
  const int l15  = lane & 15;
  const int koff = (lane >> 4) << 1;
  const int hi8  = (lane >> 4) << 3;
  __shared__ float sc[16 * 64];

  v2f qa[16];
  const float* qrow = Q + (size_t)(qt * 16 + l15) * 1024 + h * 64 + koff;
#pragma unroll
  for (int i = 0; i < 16; ++i) qa[i] = ld2(qrow + i * 4);

  const float* k1h = K1 + h * 4096;
  v8f a0 = {}, a1 = {}, a2 = {}, a3 = {};

  v2f c0 = ld2(&k1h[(0 * 16 + l15) * 64 + koff]);
  v2f c1 = ld2(&k1h[(1 * 16 + l15) * 64 + koff]);
  v2f c2 = ld2(&k1h[(2 * 16 + l15) * 64 + koff]);
  v2f c3 = ld2(&k1h[(3 * 16 + l15) * 64 + koff]);
#pragma unroll
  for (int i = 1; i < 16; ++i) {
    int kb = i * 4 + koff;
    v2f n0 = ld2(&k1h[(0 * 16 + l15) * 64 + kb]);
    v2f n1 = ld2(&k1h[(1 * 16 + l15) * 64 + kb]);
    v2f n2 = ld2(&k1h[(2 * 16 + l15) * 64 + kb]);
    v2f n3 = ld2(&k1h[(3 * 16 + l15) * 64 + kb]);
    a0 = wmma4(qa[i - 1], c0, a0);
    a1 = wmma4(qa[i - 1], c1, a1);
    a2 = wmma4(qa[i - 1], c2, a2);
    a3 = wmma4(qa[i - 1], c3, a3);
    c0 = n0; c1 = n1; c2 = n2; c3 = n3;
  }
  a0 = wmma4(qa[15], c0, a0);
  a1 = wmma4(qa[15], c1, a1);
  a2 = wmma4(qa[15], c2, a2);
  a3 = wmma4(qa[15], c3, a3);

#pragma unroll
  for (int r = 0; r < 8; ++r) {
    int row = r + hi8;
    sc[row * 64 +  0 + l15] = a0[r];
    sc[row * 64 + 16 + l15] = a1[r];
    sc[row * 64 + 32 + l15] = a2[r];
    sc[row * 64 + 48 + l15] = a3[r];
  }
  __syncthreads();
  if (lane < 16) {
    const float* row = &sc[lane * 64];
    unsigned long long taken = 0;
#pragma unroll 1
    for (int j = 0; j < 8; ++j) {
      float best = -INFINITY; int bi = 0;
      for (int cidx = 0; cidx < 64; ++cidx) {
        if ((taken >> cidx) & 1ull) continue;
        float v = row[cidx];
        if (v > best) { best = v; bi = cidx; }
      }
      taken |= (1ull << bi);
    }
    Msk[h * 2048 + qt * 16 + lane] = taken;
  }
}

// ---------------------------------------------------------------------------
// Block-sparse causal flash attention. One wave per (16-row q-tile, head).
// Iterates only 32-wide K chunks in the causal window or in the union of the
// tile's selected-chunk bitmasks. S = Q K^T and O += P V via f32 WMMA;
// P goes through LDS to convert C-layout -> A-layout.
// ---------------------------------------------------------------------------
__global__ __launch_bounds__(32) void flash_attn(const float* __restrict__ Q,
                                                 const float* __restrict__ Kb,
                                                 const float* __restrict__ Vb,
                                                 const unsigned long long* __restrict__ Msk,
                                                 float* __restrict__ O) {
  const int qt = blockIdx.x, h = blockIdx.y;
  const int lane = threadIdx.x;
  const int l15  = lane & 15;
  const int koff = (lane >> 4) << 1;
  const int hi8  = (lane >> 4) << 3;

  __shared__ float Pld[16 * 32];
  __shared__ float s_m[16], s_l[16], s_f[16];
  __shared__ unsigned long long s_rm[16];

  if (lane < 16) {
    s_rm[lane] = Msk[h * 2048 + qt * 16 + lane];
    s_m[lane] = NEG_BIG;
    s_l[lane] = 0.0f;
  }

  v2f qa[16];
  const float* qrow = Q + (size_t)(qt * 16 + l15) * 1024 + h * 64 + koff;
#pragma unroll
  for (int i = 0; i < 16; ++i) qa[i] = ld2(qrow + i * 4);

  v8f o0 = {}, o1 = {}, o2 = {}, o3 = {};

  unsigned long long uni = 0;
  for (int i = 0; i < 16; ++i) uni |= s_rm[i];

  const int row0  = qt * 16;
  const int c_hi  = (row0 + 15) >> 5;                       // causal chunk bound
  const int c_wlo = row0 >= 511 ? ((row0 - 511) >> 5) : 0;  // window chunk bound

  for (int c = 0; c <= c_hi; ++c) {
    if (c < c_wlo && !((uni >> c) & 1ull)) continue;  // chunk needed by no row

    // ---- S = Q * K_c^T : two 16x16 tiles (pipelined) ----
    v8f s0 = {}, s1 = {};
    const float* kc0 = Kb + (size_t)(c * 32 + l15) * 1024 + h * 64 + koff;
    const float* kc1 = kc0 + 16 * 1024;
    v2f kb0 = ld2(kc0);
    v2f kb1 = ld2(kc1);
#pragma unroll
    for (int i = 1; i < 16; ++i) {
      v2f n0 = ld2(kc0 + i * 4);
      v2f n1 = ld2(kc1 + i * 4);
      s0 = wmma4(qa[i - 1], kb0, s0);
      s1 = wmma4(qa[i - 1], kb1, s1);
      kb0 = n0; kb1 = n1;
    }
    s0 = wmma4(qa[15], kb0, s0);
    s1 = wmma4(qa[15], kb1, s1);

    // ---- scale + mask ----
    float p0[8], p1[8];
#pragma unroll
    for (int r = 0; r < 8; ++r) {
      int row = row0 + r + hi8;
      unsigned long long mk = s_rm[r + hi8];
      bool sel = (mk >> c) & 1ull;
      int col0 = c * 32 + l15;
      int col1 = col0 + 16;
      bool a0ok = (col0 <= row) && (sel || col0 >= row - 511);
      bool a1ok = (col1 <= row) && (sel || col1 >= row - 511);
      p0[r] = a0ok ? s0[r] * SCALE_F : NEG_BIG;
      p1[r] = a1ok ? s1[r] * SCALE_F : NEG_BIG;
    }

    // ---- online-softmax max update (reduce across the 16-lane half) ----
#pragma unroll
    for (int r = 0; r < 8; ++r) {
      float mx = fmaxf(p0[r], p1[r]);
      mx = fmaxf(mx, __shfl_xor(mx, 1, 32));
      mx = fmaxf(mx, __shfl_xor(mx, 2, 32));
      mx = fmaxf(mx, __shfl_xor(mx, 4, 32));
      mx = fmaxf(mx, __shfl_xor(mx, 8, 32));
      if (l15 == 0) {
        int row = r + hi8;
        float mo = s_m[row];
        float mn = fmaxf(mo, mx);
        s_m[row] = mn;
        s_f[row] = __expf(mo - mn);
      }
    }

    // ---- P = exp(S - m), row sums, stash P in LDS (row-major 16x32) ----
#pragma unroll
    for (int r = 0; r < 8; ++r) {
      int row = r + hi8;
      float mn = s_m[row];
      float e0 = __expf(p0[r] - mn);
      float e1 = __expf(p1[r] - mn);
      Pld[row * 32 + l15]      = e0;
      Pld[row * 32 + 16 + l15] = e1;
      float sm = e0 + e1;
      sm += __shfl_xor(sm, 1, 32);
      sm += __shfl_xor(sm, 2, 32);
      sm += __shfl_xor(sm, 4, 32);
      sm += __shfl_xor(sm, 8, 32);
      if (l15 == 0) s_l[row] = s_l[row] * s_f[row] + sm;
    }

    // ---- rescale O by exp(m_old - m_new) ----
#pragma unroll
    for (int r = 0; r < 8; ++r) {
      float f = s_f[r + hi8];
      o0[r] *= f; o1[r] *= f; o2[r] *= f; o3[r] *= f;
    }

    // ---- O += P @ V_c  (P from LDS in A-layout; V loads pipelined) ----
    const float* vbase = Vb + (size_t)(c * 32) * 1024 + h * 64;
    {
      const float* vr = vbase + (size_t)koff * 1024;
      v2f pa_c = *(const v2f*)&Pld[l15 * 32 + koff];
      v2f c0v, c1v, c2v, c3v;
      c0v.x = vr[ 0 + l15]; c0v.y = vr[1024 +  0 + l15];
      c1v.x = vr[16 + l15]; c1v.y = vr[1024 + 16 + l15];
      c2v.x = vr[32 + l15]; c2v.y = vr[1024 + 32 + l15];
      c3v.x = vr[48 + l15]; c3v.y = vr[1024 + 48 + l15];
#pragma unroll
      for (int kk = 1; kk < 8; ++kk) {
        int krow = kk * 4 + koff;
        const float* vn = vbase + (size_t)krow * 1024;
        v2f pan = *(const v2f*)&Pld[l15 * 32 + krow];
        v2f n0, n1, n2, n3;
        n0.x = vn[ 0 + l15]; n0.y = vn[1024 +  0 + l15];
        n1.x = vn[16 + l15]; n1.y = vn[1024 + 16 + l15];
        n2.x = vn[32 + l15]; n2.y = vn[1024 + 32 + l15];
        n3.x = vn[48 + l15]; n3.y = vn[1024 + 48 + l15];
        o0 = wmma4(pa_c, c0v, o0);
        o1 = wmma4(pa_c, c1v, o1);
        o2 = wmma4(pa_c, c2v, o2);
        o3 = wmma4(pa_c, c3v, o3);
        pa_c = pan; c0v = n0; c1v = n1; c2v = n2; c3v = n3;
      }
      o0 = wmma4(pa_c, c0v, o0);
      o1 = wmma4(pa_c, c1v, o1);
      o2 = wmma4(pa_c, c2v, o2);
      o3 = wmma4(pa_c, c3v, o3);
    }
  }

  // ---- epilogue: O / l ----
  float* orow = O + h * 64 + l15;
#pragma unroll
  for (int r = 0; r < 8; ++r) {
    int row = r + hi8;
    float inv = 1.0f / s_l[row];
    size_t m = (size_t)(row0 + row) * 1024;
    orow[m +  0] = o0[r] * inv;
    orow[m + 16] = o1[r] * inv;
    orow[m + 32] = o2[r] * inv;
    orow[m + 48] = o3[r] * inv;
  }
}

// ---------------------------------------------------------------------------
extern "C" void kernel_launch(void* const* d_in, const int* in_sizes, int n_in,
                              void* d_out, int out_size, void* d_ws, size_t ws_size,
                              hipStream_t stream) {
  (void)in_sizes; (void)n_in; (void)out_size; (void)ws_size;
  const float* x    = (const float*)d_in[0];
  const float* Wq   = (const float*)d_in[1];
  const float* Wk   = (const float*)d_in[2];
  const float* Wv   = (const float*)d_in[3];
  const float* Wo   = (const float*)d_in[4];
  const float* lod1 = (const float*)d_in[5];
  // d_in[6..8] (v_lod1, k_lod2, v_lod2) do not affect the output (see analysis).
  float* out = (float*)d_out;

  const size_t SM = 2048u * 1024u;
  float* qb = (float*)d_ws;        // Q, rotated      (S, H*D)
  float* kb = qb + SM;             // K, rotated
  float* vb = kb + SM;             // V
  float* ab = vb + SM;             // attention output pre-projection
  float* k1 = ab + SM;             // (H, 64, 64) LOD1 keys
  unsigned long long* mk = (unsigned long long*)(k1 + 16 * 64 * 64);  // (H,S) top8 masks

  dim3 gg(1024 / 64, 2048 / 64);
  gemm_xwt<<<gg, 128, 0, stream>>>(x, Wq, qb, 1024, 1024);
  gemm_xwt<<<gg, 128, 0, stream>>>(x, Wk, kb, 1024, 1024);
  gemm_xwt<<<gg, 128, 0, stream>>>(x, Wv, vb, 1024, 1024);
  rope_qk<<<(2048 * 16 * 32) / 256, 256, 0, stream>>>(qb, kb);
  k1_proj<<<16, 256, 0, stream>>>(kb, lod1, k1);
  lod_topk<<<dim3(128, 16), 32, 0, stream>>>(qb, k1, mk);
  flash_attn<<<dim3(128, 16), 32, 0, stream>>>(qb, kb, vb, mk, ab);
  gemm_xwt<<<gg, 128, 0, stream>>>(ab, Wo, out, 1024, 1024);
}